// GaborLayer_Bacon_82532091560569
// MI455X (gfx1250) — compile-verified
//
#include <hip/hip_runtime.h>

typedef __attribute__((ext_vector_type(2))) float v2f;
typedef __attribute__((ext_vector_type(8))) float v8f;

#define INV_2PI 0.15915494309189535f   // v_sin_f32 computes sin(2*pi*x)
#define LOG2E   1.4426950408889634f    // v_exp_f32 computes 2^x

#define ROW_TILES_PER_BLOCK 8

// out[i,o] = sin(x_i . w_o + b_o) * exp(-0.5*gamma_o*|x_i - mu_o|^2)
// Both args are linear in f(x) = [x0, x1, x0^2+x1^2, 1] (K=4) -> one
// V_WMMA_F32_16X16X4_F32 each per 16x16 tile. The 1/(2*pi) (for v_sin_f32)
// and log2(e) (for v_exp_f32) scalings are folded into the B coefficients,
// so the WMMA output feeds the TRANS ops directly with zero extra VALU.
// Each wave owns ONE column tile (B built once, registers only) and loops
// over row tiles; 16 waves per 512-thread block cover all 256 columns.
__global__ __launch_bounds__(512) void GaborLayer_wmma_kernel(
    const float* __restrict__ x,      // (B,2)
    const float* __restrict__ W,      // (256,2)
    const float* __restrict__ bias,   // (256,)
    const float* __restrict__ mu,     // (1,256,2)
    const float* __restrict__ gamma,  // (256,)
    float* __restrict__ out,          // (B,256)
    int nRowTiles)
{
    const int lane   = threadIdx.x & 31;
    const int laneLo = lane & 15;
    const bool hiHalf = lane >= 16;
    const int colTile = threadIdx.x >> 5;        // wave id 0..15 -> column tile
    const int o = colTile * 16 + laneLo;         // output column 0..255

    // ---- B tiles (4x16 f32, 2 VGPRs each): lanes 0-15 = rows {K0,K1},
    //      lanes 16-31 = rows {K2,K3}; column N = lane&15. Built once.
    const float2 wv  = ((const float2*)W)[o];
    const float  bv  = bias[o];
    const float2 muv = ((const float2*)mu)[o];
    const float  lg  = LOG2E * gamma[o];

    v2f Bs;  // sin-arg coeffs * (1/2pi): [W0, W1, 0, b] / (2*pi)
    Bs[0] = hiHalf ? 0.0f               : (INV_2PI * wv.x);
    Bs[1] = hiHalf ? (INV_2PI * bv)     : (INV_2PI * wv.y);

    v2f Be;  // exp-arg coeffs * log2(e): [g*mu0, g*mu1, -0.5g, -0.5g*|mu|^2]
    Be[0] = hiHalf ? (-0.5f * lg)                                    : (lg * muv.x);
    Be[1] = hiHalf ? (-0.5f * lg * (muv.x * muv.x + muv.y * muv.y))  : (lg * muv.y);

    for (int i = 0; i < ROW_TILES_PER_BLOCK; ++i) {
        const int tile = blockIdx.x * ROW_TILES_PER_BLOCK + i;
        if (tile >= nRowTiles) return;           // wave-uniform: EXEC stays all-1s
        const int rowBase = tile * 16;

        // ---- A matrix (16x4 f32, 2 VGPRs): lanes 0-15 hold {K0,K1} of row
        //      M=lane, lanes 16-31 hold {K2,K3} of row M=lane-16.
        const float2 xm = ((const float2*)x)[rowBase + laneLo];
        v2f A;
        A[0] = hiHalf ? (xm.x * xm.x + xm.y * xm.y) : xm.x;  // K2=|x|^2 | K0=x0
        A[1] = hiHalf ? 1.0f                        : xm.y;  // K3=1     | K1=x1

        v8f lin = {};
        v8f arg = {};
        lin = __builtin_amdgcn_wmma_f32_16x16x4_f32(
                  false, A, false, Bs, (short)0, lin, false, false);
        arg = __builtin_amdgcn_wmma_f32_16x16x4_f32(
                  false, A, false, Be, (short)0, arg, false, false);

        // ---- D layout: VGPR r -> row M = r + (hiHalf?8:0), col N = lane&15.
        // v_sin_f32 / v_exp_f32 consume the WMMA outputs directly (scalings
        // pre-folded). Each half-wave stores one aligned 64-B block per r;
        // non-temporal: 268 MB streamed, zero reuse.
        float* outRow = out + (size_t)(rowBase + (hiHalf ? 8 : 0)) * 256
                            + colTile * 16 + laneLo;
        #pragma unroll
        for (int r = 0; r < 8; ++r) {
            const float v = __builtin_amdgcn_sinf(lin[r])
                          * __builtin_amdgcn_exp2f(arg[r]);
            __builtin_nontemporal_store(v, outRow + (size_t)r * 256);
        }
    }
}

extern "C" void kernel_launch(void* const* d_in, const int* in_sizes, int n_in,
                              void* d_out, int out_size, void* d_ws, size_t ws_size,
                              hipStream_t stream) {
    const float* x     = (const float*)d_in[0];  // (B,2)
    const float* W     = (const float*)d_in[1];  // (256,2)
    const float* bias  = (const float*)d_in[2];  // (256,)
    const float* mu    = (const float*)d_in[3];  // (1,256,2)
    const float* gamma = (const float*)d_in[4];  // (256,)
    float* out = (float*)d_out;

    const int B = in_sizes[0] / 2;                       // 262144
    const int nRowTiles = (B + 15) / 16;                 // 16384
    const int nBlocks = (nRowTiles + ROW_TILES_PER_BLOCK - 1) / ROW_TILES_PER_BLOCK;

    GaborLayer_wmma_kernel<<<nBlocks, 512, 0, stream>>>(x, W, bias, mu, gamma, out,
                                                        nRowTiles);
}